// SpikingInferiorOlive_88149908783722
// MI455X (gfx1250) — compile-verified
//
#include <hip/hip_runtime.h>
#include <math.h>

// CDNA5 / gfx1250, wave32. Single-wave kernel: the whole problem is
// launch-latency bound (~1.2 kFLOP, ~0.5 KB traffic), so one workgroup of 32
// threads does everything in registers. The linear input-current stage
// I[8] = W[8x4] * f[4] + bias[8] is mapped onto one v_wmma_f32_16x16x4_f32,
// which also distributes I[n] into lane n's accumulator (D row-0 layout).

typedef __attribute__((ext_vector_type(2))) float v2f;
typedef __attribute__((ext_vector_type(8))) float v8f;

__global__ __launch_bounds__(32)
void SpikingInferiorOlive_kernel(const float* __restrict__ sens,
                                 const float* __restrict__ mot,
                                 const float* __restrict__ cpe_in,
                                 const float* __restrict__ prev_s,
                                 const float* __restrict__ prev_m,
                                 const float* __restrict__ phase_in,
                                 const float* __restrict__ v0,
                                 const float* __restrict__ u0,
                                 const float* __restrict__ rate0,
                                 const float* __restrict__ noise,
                                 const float* __restrict__ pc_m0,
                                 float* __restrict__ out) {
    const int lane = threadIdx.x;

    // Prefetch the noise table (global_prefetch_b8) while we do scalar work.
    __builtin_prefetch(noise, 0, 1);

    // ---- scalar prologue (all lanes redundantly; broadcast loads) ----
    const float s   = sens[0];
    const float m   = mot[0];
    const float cpe = cpe_in[0];

    const float dsen = fabsf(s - prev_s[0]);
    const float se   = fminf(1.0f, dsen + 0.3f * s);
    const float dmot = fabsf(m - prev_m[0]);
    const float me   = fminf(1.0f, dmot + 0.3f * m);

    const float phase = phase_in[0] + 0.04f;
    const float osc   = 0.15f * sinf(6.28318530717958647692f * phase);

    const float raw_cf = 0.4f * se + 0.4f * me + 0.2f * cpe;
    const bool  is_cs  = raw_cf > 0.2f;
    const float cf     = is_cs ? fminf(1.0f, raw_cf * 1.5f)
                               : fmaxf(0.0f, raw_cf + osc);

    // ---- WMMA: I[n] = sum_k f[k]*W[n][k] + bias[n] via 16x16x4 f32 ----
    // A (16x4): row M=0 holds f = [se, dsen, me, dmot].
    //   ISA layout: lanes 0-15 -> (K=0 in v0, K=1 in v1), lanes 16-31 -> (K=2, K=3),
    //   with M = lane%16. So only lane 0 (K=0,1) and lane 16 (K=2,3) are nonzero.
    const bool l0  = (lane == 0);
    const bool l16 = (lane == 16);
    v2f a;
    a.x = l0 ? se   : (l16 ? me   : 0.0f);   // K=0 / K=2
    a.y = l0 ? dsen : (l16 ? dmot : 0.0f);   // K=1 / K=3
    // B (4x16) = W^T: B[k][n] = W[n][k]. Layout mirrors A:
    //   K = 2*(lane>=16) + vgpr, N = lane%16.
    const int  n  = lane & 15;
    const bool hi = lane >= 16;
    v2f b;
    b.x = hi ? ((n == 4) ? 15.0f : ((n == 6) ? 8.0f : 0.0f))     // K=2: W[n][2]
             : ((n == 0) ? 15.0f : ((n == 2) ? 8.0f : 0.0f));    // K=0: W[n][0]
    b.y = hi ? ((n == 5) ? 12.0f : 0.0f)                         // K=3: W[n][3]
             : ((n == 1) ? 12.0f : 0.0f);                        // K=1: W[n][1]
    // C (16x16): row 0 carries bias[n]: neuron 3 -> 5*osc+2, neuron 7 -> 10*cf.
    v8f c = {};
    c[0] = (lane == 3) ? (5.0f * osc + 2.0f)
         : ((lane == 7) ? (10.0f * cf) : 0.0f);

    // (neg_a, A, neg_b, B, c_mod, C, reuse_a, reuse_b) -- EXEC is all-1s here.
    v8f d = __builtin_amdgcn_wmma_f32_16x16x4_f32(
        false, a, false, b, (short)0, c, false, false);

    // D row 0: VGPR0, lanes 0-15 hold D[0][lane] => lane j<8 holds I[j].
    const float Icur = d[0];

    // ---- Izhikevich scan: lane j simulates neuron j for 10 steps ----
    const int j = lane & 7;               // lanes 8..31 shadow-compute harmlessly
    float v = v0[j];
    float u = u0[j];
    float r = rate0[j];
#pragma unroll
    for (int t = 0; t < 10; ++t) {
        const float nz = noise[t * 8 + j];
        const float Ii = Icur + nz * 0.3f - 1.5f;          // I_TONIC = -1.5
        const float dv = 0.04f * v * v + 5.0f * v + 140.0f - u + Ii;
        const float du = 0.02f * (0.2f * v - u);
        v += dv;
        u += du;
        const bool sp = (v >= 30.0f);
        v = sp ? -65.0f : v;
        u = sp ? (u + 8.0f) : u;
        r = 0.95f * r + 0.05f * (sp ? 1.0f : 0.0f);
    }

    // ---- predictive-coding column: 8 leaky-integrator substeps ----
    const float drive = (j < 4) ? se : me;                 // prediction == 0
    float mem = pc_m0[j];
#pragma unroll
    for (int t = 0; t < 8; ++t) mem += 0.125f * (drive - mem);

    // ---- reductions via LDS ----
    __shared__ float s_r[8];
    __shared__ float s_mem[8];
    if (lane < 8) {
        s_r[lane]   = r;
        s_mem[lane] = mem;
        out[8 + lane] = r;                                 // rate[0..7]
    }
    __syncthreads();

    if (lane == 0) {
        const float pe0 = 0.25f * (s_mem[0] + s_mem[1] + s_mem[2] + s_mem[3]);
        const float pe1 = 0.25f * (s_mem[4] + s_mem[5] + s_mem[6] + s_mem[7]);
        const float p0  = 1.0f / (1.0f + pe0 * pe0);
        const float p1  = 1.0f / (1.0f + pe1 * pe1);
        const float free_energy = 0.5f * (p0 * pe0 * pe0 + p1 * pe1 * pe1);
        const float pred_err    = 0.5f * (fabsf(pe0) + fabsf(pe1));
        const float rsum = s_r[0] + s_r[1] + s_r[2] + s_r[3]
                         + s_r[4] + s_r[5] + s_r[6] + s_r[7];

        out[0] = se;
        out[1] = me;
        out[2] = cf;
        out[3] = is_cs ? 1.0f : 0.0f;
        out[4] = pred_err;
        out[5] = 0.5f * (p0 + p1);
        out[6] = free_energy;
        out[7] = rsum * 0.125f;
    }
}

extern "C" void kernel_launch(void* const* d_in, const int* in_sizes, int n_in,
                              void* d_out, int out_size, void* d_ws, size_t ws_size,
                              hipStream_t stream) {
    (void)in_sizes; (void)n_in; (void)out_size; (void)d_ws; (void)ws_size;
    const float* sens    = (const float*)d_in[0];   // sensory_surprise (1)
    const float* mot     = (const float*)d_in[1];   // motor_mismatch   (1)
    const float* cpe     = (const float*)d_in[2];   // cerebellar_pe    (1)
    const float* prev_s  = (const float*)d_in[3];   // prev_sensory     (1)
    const float* prev_m  = (const float*)d_in[4];   // prev_motor       (1)
    const float* phase   = (const float*)d_in[5];   // osc_phase        (1)
    const float* v0      = (const float*)d_in[6];   // v0               (8)
    const float* u0      = (const float*)d_in[7];   // u0               (8)
    const float* rate0   = (const float*)d_in[8];   // rate0            (8)
    const float* noise   = (const float*)d_in[9];   // noise         (10,8)
    const float* pc_m0   = (const float*)d_in[10];  // pc_m0            (8)
    float* out = (float*)d_out;                     // 16 floats

    SpikingInferiorOlive_kernel<<<1, 32, 0, stream>>>(
        sens, mot, cpe, prev_s, prev_m, phase, v0, u0, rate0, noise, pc_m0, out);
}